// EquivariantProductBasisBlock_46531675685335
// MI455X (gfx1250) — compile-verified
//
#include <hip/hip_runtime.h>
#include <hip/hip_bf16.h>

typedef __attribute__((ext_vector_type(2))) float v2f;
typedef __attribute__((ext_vector_type(8))) float v8f;

#define ST 132          // padded LDS row stride (132 mod 64 = 4 -> conflict-free A-frag reads)
#define NSPEC 64

// ---------------------------------------------------------------- planning ---
__global__ void k_zero(int* p, int n) {
    int i = blockIdx.x * blockDim.x + threadIdx.x;
    if (i < n) p[i] = 0;
}

__global__ void k_count(const int* __restrict__ species, int* __restrict__ counts, int N) {
    int i = blockIdx.x * blockDim.x + threadIdx.x;
    if (i < N) atomicAdd(&counts[species[i]], 1);
}

// single block, 64 threads: prefix sums + block table + cursors
__global__ void k_plan(const int* __restrict__ counts,
                       int* __restrict__ cursors,
                       int* __restrict__ nblocks,
                       int* __restrict__ bspecies,
                       int* __restrict__ bstart,
                       int* __restrict__ bcount) {
    __shared__ int s_cnt[NSPEC], s_start[NSPEC], s_bbase[NSPEC];
    int t = threadIdx.x;
    if (t < NSPEC) s_cnt[t] = counts[t];
    __syncthreads();
    if (t == 0) {
        int start = 0, bb = 0;
        for (int s = 0; s < NSPEC; ++s) {
            s_start[s] = start;  start += s_cnt[s];
            s_bbase[s] = bb;     bb += (s_cnt[s] + 15) >> 4;
        }
        *nblocks = bb;
    }
    __syncthreads();
    if (t < NSPEC) {
        int c = s_cnt[t], st = s_start[t], bb = s_bbase[t];
        cursors[t] = st;
        int nb = (c + 15) >> 4;
        for (int i = 0; i < nb; ++i) {
            bspecies[bb + i] = t;
            bstart[bb + i]   = st + 16 * i;
            int rem = c - 16 * i;
            bcount[bb + i]   = rem < 16 ? rem : 16;
        }
    }
}

__global__ void k_scatter(const int* __restrict__ species, int* __restrict__ cursors,
                          int* __restrict__ perm, int N) {
    int i = blockIdx.x * blockDim.x + threadIdx.x;
    if (i < N) {
        int pos = atomicAdd(&cursors[species[i]], 1);
        perm[pos] = i;
    }
}

// ---------------------------------------------------------------- main -------
__global__ __launch_bounds__(256)
void eqprod_main(const float* __restrict__ node_feats,
                 const float* __restrict__ w0,
                 const float* __restrict__ w1,
                 const float* __restrict__ gate_kernel,
                 const float* __restrict__ gate_bias,
                 const float* __restrict__ lin_w0,
                 const float* __restrict__ lin_w1,
                 const int*   __restrict__ nblocks,
                 const int*   __restrict__ bspecies,
                 const int*   __restrict__ bstart,
                 const int*   __restrict__ bcount,
                 const int*   __restrict__ perm,
                 float*       __restrict__ out)
{
    const int b = blockIdx.x;
    if (b >= *nblocks) return;

    __shared__ float s_out0[16 * ST];   // out0: [node][f]
    __shared__ float s_out1[48 * ST];   // out1: [node*3 + d][f]
    __shared__ int   s_node[16];

    const int tid   = threadIdx.x;
    const int sp    = bspecies[b];      // uniform per block (species-sorted)
    const int start = bstart[b];
    const int cnt   = bcount[b];

    if (tid < 16) s_node[tid] = perm[start + (tid < cnt ? tid : 0)];
    __syncthreads();

    // ------------- Stage 1: paths0/paths1 * per-species weights --------------
    #pragma unroll
    for (int it = 0; it < 8; ++it) {
        int idx = tid + it * 256;
        int n = idx >> 7;
        int f = idx & 127;
        int gn = s_node[n];

        float x0 = node_feats[(size_t)gn * 512 + f];
        const float* x1p = node_feats + (size_t)gn * 512 + 128 + f * 3;
        float xa = x1p[0], xb = x1p[1], xc = x1p[2];
        float dot = xa * xa + xb * xb + xc * xc;

        const float* w0p = w0 + sp * (5 * 128) + f;
        const float* w1p = w1 + sp * (4 * 128) + f;
        float x0sq = x0 * x0;
        float o0 = x0        * w0p[0]
                 + x0sq      * w0p[128]
                 + dot       * w0p[256]
                 + x0sq * x0 * w0p[384]
                 + x0 * dot  * w0p[512];
        float c1 = w1p[0] + x0 * w1p[128] + x0sq * w1p[256] + dot * w1p[384];

        s_out0[n * ST + f]           = o0;
        s_out1[(n * 3 + 0) * ST + f] = xa * c1;
        s_out1[(n * 3 + 1) * ST + f] = xb * c1;
        s_out1[(n * 3 + 2) * ST + f] = xc * c1;
    }
    __syncthreads();

    const int lane  = tid & 31;
    const int wave  = tid >> 5;
    const int mrow  = lane & 15;
    const int khalf = lane >> 4;

    // ------------- Stage 2: gate = out0 @ gk[sp] + gb[sp] via WMMA -----------
    // 16 col tiles of 16x16 over K=128; wave handles cols [wave*16..] (o<128)
    // and [wave*16+128..] (o>=128). Bias folded into the accumulator init.
    {
        const float* gk = gate_kernel + (size_t)sp * (128 * 256);
        const int colA = wave * 16 + mrow;        // gate col < 128  -> scales out0
        const int colB = colA + 128;              // gate col >= 128 -> scales out1
        float bias0 = gate_bias[sp * 256 + colA];
        float bias1 = gate_bias[sp * 256 + colB];
        v8f g0, g1;
        #pragma unroll
        for (int r = 0; r < 8; ++r) { g0[r] = bias0; g1[r] = bias1; }

        const float* Arow = s_out0 + mrow * ST + 2 * khalf;
        #pragma unroll 4
        for (int kb = 0; kb < 128; kb += 4) {
            v2f a = *(const v2f*)(Arow + kb);     // shared A-frag for both tiles
            int k0 = kb + 2 * khalf;
            v2f b0; b0.x = gk[k0 * 256 + colA]; b0.y = gk[(k0 + 1) * 256 + colA];
            v2f b1; b1.x = gk[k0 * 256 + colB]; b1.y = gk[(k0 + 1) * 256 + colB];
            g0 = __builtin_amdgcn_wmma_f32_16x16x4_f32(false, a, false, b0, (short)0, g0, false, false);
            g1 = __builtin_amdgcn_wmma_f32_16x16x4_f32(false, a, false, b1, (short)0, g1, false, false);
        }
        __syncthreads();   // all gate reads of s_out0 complete before scaling

        #pragma unroll
        for (int r = 0; r < 8; ++r) {
            int M = r + 8 * khalf;                // D layout: VGPR r -> row r + 8*khalf
            s_out0[M * ST + colA]           *= g0[r];
            s_out1[(M * 3 + 0) * ST + colA] *= g1[r];   // f = colB-128 = colA
            s_out1[(M * 3 + 1) * ST + colA] *= g1[r];
            s_out1[(M * 3 + 2) * ST + colA] *= g1[r];
        }
    }
    __syncthreads();

    // ------------- Stage 3: y0 / y1 linear layers via WMMA -------------------
    // Wave owns output col tile ct = wave; fuses y0 + 3 y1 row-tiles in one
    // K-loop so the lin_w1 B-fragment is loaded once for 3 WMMAs.
    {
        const float inv = 0.08838834764831845f;   // 1/sqrt(128)
        const int col = wave * 16 + mrow;
        const float* B0 = lin_w0 + col;
        const float* B1 = lin_w1 + col;
        const float* A0 = s_out0 + mrow * ST + 2 * khalf;
        const float* A1 = s_out1 + (     mrow) * ST + 2 * khalf;
        const float* A2 = s_out1 + (16 + mrow) * ST + 2 * khalf;
        const float* A3 = s_out1 + (32 + mrow) * ST + 2 * khalf;

        v8f acc0 = {}, acc1 = {}, acc2 = {}, acc3 = {};
        #pragma unroll 4
        for (int kb = 0; kb < 128; kb += 4) {
            int k0 = kb + 2 * khalf;
            v2f bw0; bw0.x = B0[k0 * 128]; bw0.y = B0[(k0 + 1) * 128];
            v2f bw1; bw1.x = B1[k0 * 128]; bw1.y = B1[(k0 + 1) * 128];
            v2f a0 = *(const v2f*)(A0 + kb);
            v2f a1 = *(const v2f*)(A1 + kb);
            v2f a2 = *(const v2f*)(A2 + kb);
            v2f a3 = *(const v2f*)(A3 + kb);
            acc0 = __builtin_amdgcn_wmma_f32_16x16x4_f32(false, a0, false, bw0, (short)0, acc0, false, false);
            acc1 = __builtin_amdgcn_wmma_f32_16x16x4_f32(false, a1, false, bw1, (short)0, acc1, false, false);
            acc2 = __builtin_amdgcn_wmma_f32_16x16x4_f32(false, a2, false, bw1, (short)0, acc2, false, false);
            acc3 = __builtin_amdgcn_wmma_f32_16x16x4_f32(false, a3, false, bw1, (short)0, acc3, false, false);
        }

        #pragma unroll
        for (int r = 0; r < 8; ++r) {             // y0
            int M = r + 8 * khalf;
            if (M < cnt)
                out[(size_t)s_node[M] * 512 + col] = acc0[r] * inv;
        }
        #pragma unroll
        for (int r = 0; r < 8; ++r) {             // y1 row-tile 0: rows 0..15
            int M = r + 8 * khalf;
            int n = M / 3, d = M - 3 * n;
            if (n < cnt)
                out[(size_t)s_node[n] * 512 + 128 + 3 * col + d] = acc1[r] * inv;
        }
        #pragma unroll
        for (int r = 0; r < 8; ++r) {             // y1 row-tile 1: rows 16..31
            int M = 16 + r + 8 * khalf;
            int n = M / 3, d = M - 3 * n;
            if (n < cnt)
                out[(size_t)s_node[n] * 512 + 128 + 3 * col + d] = acc2[r] * inv;
        }
        #pragma unroll
        for (int r = 0; r < 8; ++r) {             // y1 row-tile 2: rows 32..47
            int M = 32 + r + 8 * khalf;
            int n = M / 3, d = M - 3 * n;
            if (n < cnt)
                out[(size_t)s_node[n] * 512 + 128 + 3 * col + d] = acc3[r] * inv;
        }
    }
}

// ---------------------------------------------------------------- launch -----
extern "C" void kernel_launch(void* const* d_in, const int* in_sizes, int n_in,
                              void* d_out, int out_size, void* d_ws, size_t ws_size,
                              hipStream_t stream) {
    const float* node_feats   = (const float*)d_in[0];
    const int*   node_species = (const int*)  d_in[1];
    const float* w0           = (const float*)d_in[2];
    const float* w1           = (const float*)d_in[3];
    const float* gate_kernel  = (const float*)d_in[4];
    const float* gate_bias    = (const float*)d_in[5];
    const float* lin_w0       = (const float*)d_in[6];
    const float* lin_w1       = (const float*)d_in[7];
    float*       out          = (float*)d_out;

    const int N    = in_sizes[1];
    const int maxB = (N + 15) / 16 + NSPEC;   // worst case: one partial block per species

    // workspace layout (ints)
    int* counts   = (int*)d_ws;               // 64
    int* cursors  = counts  + NSPEC;          // 64
    int* nblocks  = cursors + NSPEC;          // 1
    int* bspecies = nblocks + 1;              // maxB
    int* bstart   = bspecies + maxB;          // maxB
    int* bcount   = bstart   + maxB;          // maxB
    int* perm     = bcount   + maxB;          // N

    k_zero<<<1, 256, 0, stream>>>(counts, 2 * NSPEC + 1);
    k_count<<<(N + 255) / 256, 256, 0, stream>>>(node_species, counts, N);
    k_plan<<<1, 64, 0, stream>>>(counts, cursors, nblocks, bspecies, bstart, bcount);
    k_scatter<<<(N + 255) / 256, 256, 0, stream>>>(node_species, cursors, perm, N);
    eqprod_main<<<maxB, 256, 0, stream>>>(node_feats, w0, w1, gate_kernel, gate_bias,
                                          lin_w0, lin_w1, nblocks, bspecies, bstart,
                                          bcount, perm, out);
}